// FNO_RCLN_v3_44341242363998
// MI455X (gfx1250) — compile-verified
//
#include <hip/hip_runtime.h>
#include <math.h>

// ============================================================================
// FNO3d forward for MI455X (gfx1250).
// Strategy: truncated separable DFT (only 8x8x4 modes survive), so all FFT
// work becomes small dense GEMMs. The dominant one (W-dim forward DFT,
// [64x64]x[64x8] per (b,c,d)) runs on V_WMMA_F32_16X16X4_F32. Everything is
// L2-resident (activation = 67MB < 192MB L2). In-place layer update.
// ============================================================================

typedef float v2f __attribute__((ext_vector_type(2)));
typedef float v8f __attribute__((ext_vector_type(8)));

#define TWO_PI_OVER_64 0.09817477042468103f

// ---- workspace layout (float offsets) ----
// tables
static const size_t OFF_BTAB = 0;                    // [16 kstep][2 vgpr][32 lane] = 1024
static const size_t OFF_FH   = 1024;                 // fwd DFT  [8 kidx][64][2]    = 1024
static const size_t OFF_IF   = 2048;                 // inv DFT  [8 kidx][64][2]/64 = 1024
static const size_t OFF_IW   = 3072;                 // irfft-W  [64 w][4 k][2]     = 512
// activation x: [8 b][8 c][64][64][64]
static const size_t OFF_X    = 4096;                 // 16,777,216 floats (64 MB)
// spectral scratch (Yh aliases Xw, Yd aliases Xh; lifetimes disjoint)
static const size_t OFF_XWRE = OFF_X    + 16777216;  // [b*c][64 d][4 kw][64 h] = 1,048,576
static const size_t OFF_XWIM = OFF_XWRE + 1048576;
static const size_t OFF_XHRE = OFF_XWIM + 1048576;   // [b*c][8 kh][4 kw][64 d] = 131,072
static const size_t OFF_XHIM = OFF_XHRE + 131072;
static const size_t OFF_ZTRE = OFF_XHIM + 131072;    // [b][c][8 kd][8 kh][4 kw] = 16,384
static const size_t OFF_ZTIM = OFF_ZTRE + 16384;
static const size_t OFF_FTRE = OFF_ZTIM + 16384;     // mixed modes, same shape (o-channel)
static const size_t OFF_FTIM = OFF_FTRE + 16384;
// total ~19.2M floats = ~77 MB

__device__ __forceinline__ int modek(int i) { return (i < 4) ? i : (56 + i); } // {0..3,60..63}

__device__ __forceinline__ float gelu_tanh(float v) {
  float v3 = v * v * v;
  return 0.5f * v * (1.0f + tanhf(0.7978845608028654f * (v + 0.044715f * v3)));
}

// ---------------------------------------------------------------------------
// Table builder (3584 threads total).
// ---------------------------------------------------------------------------
__global__ void k_tables(float* __restrict__ ws) {
  int t = blockIdx.x * blockDim.x + threadIdx.x;
  if (t < 1024) {
    // B-fragments for the fwd-W WMMA GEMM. Column n<4: cos(2pi*n*w/64);
    // n in 4..7: sin(2pi*(n-4)*w/64) (negated at store time for the -i part).
    // Fragment layout mirrors the ISA A/C striping: vgpr r, lanes>=16 -> K+2.
    int lane = t & 31, r = (t >> 5) & 1, k = t >> 6;
    int n = lane & 15;
    int w = 4 * k + ((lane >> 4) << 1) + r;
    float v = 0.0f;
    if (n < 4)      v = cosf(TWO_PI_OVER_64 * (float)(n * w));
    else if (n < 8) v = sinf(TWO_PI_OVER_64 * (float)((n - 4) * w));
    ws[OFF_BTAB + t] = v;
  } else if (t < 2048) {
    int i = t - 1024;                 // (kidx*64 + h)*2 + c
    int c = i & 1, h = (i >> 1) & 63, kidx = i >> 7;
    float th = TWO_PI_OVER_64 * (float)(modek(kidx) * h);
    ws[OFF_FH + i] = (c == 0) ? cosf(th) : -sinf(th);     // e^{-i th}
  } else if (t < 3072) {
    int i = t - 2048;
    int c = i & 1, h = (i >> 1) & 63, kidx = i >> 7;
    float th = TWO_PI_OVER_64 * (float)(modek(kidx) * h);
    ws[OFF_IF + i] = ((c == 0) ? cosf(th) : sinf(th)) * (1.0f / 64.0f); // e^{+i th}/64
  } else if (t < 3584) {
    int i = t - 3072;                 // (w*4 + k)*2 + c
    int c = i & 1, k = (i >> 1) & 3, w = i >> 3;
    float th = TWO_PI_OVER_64 * (float)(k * w);
    float amp = ((k == 0) ? 1.0f : 2.0f) * (1.0f / 64.0f);
    ws[OFF_IW + i] = ((c == 0) ? cosf(th) : -sinf(th)) * amp;
  }
}

// ---------------------------------------------------------------------------
// Lift: x[b,o,p] = sum_c fc0_w[o,c]*u[b,c,p] + fc0_b[o]
// ---------------------------------------------------------------------------
__global__ __launch_bounds__(256) void k_fc0(const float* __restrict__ u,
                                             const float* __restrict__ w,
                                             const float* __restrict__ b,
                                             float* __restrict__ x) {
  int t = blockIdx.x * blockDim.x + threadIdx.x;   // b*262144 + p
  int p = t & 262143, bb = t >> 18;
  float u0 = u[(size_t)(bb * 3 + 0) * 262144 + p];
  float u1 = u[(size_t)(bb * 3 + 1) * 262144 + p];
  float u2 = u[(size_t)(bb * 3 + 2) * 262144 + p];
#pragma unroll
  for (int o = 0; o < 8; ++o)
    x[(size_t)(bb * 8 + o) * 262144 + p] =
        w[o * 3] * u0 + w[o * 3 + 1] * u1 + w[o * 3 + 2] * u2 + b[o];
}

// ---------------------------------------------------------------------------
// Forward rDFT along W via WMMA: per (b,c,d): [64h x 64w] * [64w x 8].
// Cols 0..3 -> Re(kw), cols 4..7 -> -Im source (sin), 8..15 zero pad.
// One block = one (b,c,d); 4 waves, wave m owns h rows [16m, 16m+16).
// ---------------------------------------------------------------------------
__global__ __launch_bounds__(128) void k_fwd_w(const float* __restrict__ x,
                                               float* __restrict__ xwre,
                                               float* __restrict__ xwim,
                                               const float* __restrict__ btab) {
  const int bcd = blockIdx.x;                 // (b*8+c)*64 + d
  const int bc = bcd >> 6, d = bcd & 63;
  const int lane = threadIdx.x & 31;
  const int wave = threadIdx.x >> 5;
  const int hbase = wave * 16;
  const float* xrow = x + (size_t)bc * 262144 + (size_t)d * 4096;  // [64h][64w]

  // Preload all 16 B fragments (uniform across blocks; L2-resident table).
  v2f bfrag[16];
#pragma unroll
  for (int k = 0; k < 16; ++k) {
    bfrag[k].x = btab[(k * 2 + 0) * 32 + lane];
    bfrag[k].y = btab[(k * 2 + 1) * 32 + lane];
  }

  const int m = lane & 15;
  const int khalf = (lane >> 4) << 1;         // lanes>=16 carry K+2
  v8f acc = {};
#pragma unroll
  for (int k = 0; k < 16; ++k) {
    const float* ap = xrow + (hbase + m) * 64 + 4 * k + khalf;
    v2f a; a.x = ap[0]; a.y = ap[1];
    acc = __builtin_amdgcn_wmma_f32_16x16x4_f32(
        /*neg_a=*/false, a, /*neg_b=*/false, bfrag[k],
        /*c_mod=*/(short)0, acc, /*reuse_a=*/false, /*reuse_b=*/false);
  }

  // C layout: vgpr r = row hbase+r (lanes<16) / hbase+8+r (lanes>=16), col = lane&15.
  if (m < 8) {
    int kw = m & 3;
    int h0 = hbase + ((lane >> 4) << 3);
    float sgn = (m < 4) ? 1.0f : -1.0f;       // Im = -sum(x*sin)
    float* dst = ((m < 4) ? xwre : xwim) + ((size_t)bcd * 4 + kw) * 64 + h0;
    float4 lo, hi;
    lo.x = sgn * acc[0]; lo.y = sgn * acc[1]; lo.z = sgn * acc[2]; lo.w = sgn * acc[3];
    hi.x = sgn * acc[4]; hi.y = sgn * acc[5]; hi.z = sgn * acc[6]; hi.w = sgn * acc[7];
    *(float4*)(dst) = lo;
    *(float4*)(dst + 4) = hi;
  }
}

// ---------------------------------------------------------------------------
// Forward DFT along H (complex*complex, 8 modes). 131072 threads.
// ---------------------------------------------------------------------------
__global__ __launch_bounds__(256) void k_fwd_h(const float* __restrict__ xwre,
                                               const float* __restrict__ xwim,
                                               const float* __restrict__ fh,
                                               float* __restrict__ xhre,
                                               float* __restrict__ xhim) {
  int t = blockIdx.x * blockDim.x + threadIdx.x;
  int d = t & 63, kw = (t >> 6) & 3, kh = (t >> 8) & 7, bc = t >> 11;
  const float* re = xwre + ((size_t)(bc * 64 + d) * 4 + kw) * 64;
  const float* im = xwim + ((size_t)(bc * 64 + d) * 4 + kw) * 64;
  const float* f = fh + kh * 128;
  float ar = 0.0f, ai = 0.0f;
  for (int h = 0; h < 64; ++h) {
    float cr = f[2 * h], ci = f[2 * h + 1];
    float xr = re[h], xi = im[h];
    ar += cr * xr - ci * xi;
    ai += cr * xi + ci * xr;
  }
  size_t o = (((size_t)bc * 8 + kh) * 4 + kw) * 64 + d;
  xhre[o] = ar; xhim[o] = ai;
}

// ---------------------------------------------------------------------------
// Forward DFT along D (8 modes). 16384 threads.
// ---------------------------------------------------------------------------
__global__ __launch_bounds__(256) void k_fwd_d(const float* __restrict__ xhre,
                                               const float* __restrict__ xhim,
                                               const float* __restrict__ fh,
                                               float* __restrict__ ztre,
                                               float* __restrict__ ztim) {
  int t = blockIdx.x * blockDim.x + threadIdx.x;
  int kw = t & 3, kh = (t >> 2) & 7, kd = (t >> 5) & 7, bc = t >> 8;
  const float* re = xhre + (((size_t)bc * 8 + kh) * 4 + kw) * 64;
  const float* im = xhim + (((size_t)bc * 8 + kh) * 4 + kw) * 64;
  const float* f = fh + kd * 128;
  float ar = 0.0f, ai = 0.0f;
  for (int d = 0; d < 64; ++d) {
    float cr = f[2 * d], ci = f[2 * d + 1];
    float xr = re[d], xi = im[d];
    ar += cr * xr - ci * xi;
    ai += cr * xi + ci * xr;
  }
  size_t o = (((size_t)bc * 8 + kd) * 8 + kh) * 4 + kw;
  ztre[o] = ar; ztim[o] = ai;
}

// ---------------------------------------------------------------------------
// Mode mixing: out[b,o,kd,kh,kw] = sum_i z[b,i,...] * w_oct[i,o,kx,ky,kz]
// ---------------------------------------------------------------------------
__global__ __launch_bounds__(256) void k_mix(const float* __restrict__ ztre,
                                             const float* __restrict__ ztim,
                                             const float* __restrict__ w1,
                                             const float* __restrict__ w2,
                                             const float* __restrict__ w3,
                                             const float* __restrict__ w4,
                                             float* __restrict__ ftre,
                                             float* __restrict__ ftim) {
  int t = blockIdx.x * blockDim.x + threadIdx.x;
  int kw = t & 3, kh = (t >> 2) & 7, kd = (t >> 5) & 7, o = (t >> 8) & 7, b = t >> 11;
  const float* w = (kd < 4) ? ((kh < 4) ? w1 : w3) : ((kh < 4) ? w2 : w4);
  int kx = kd & 3, ky = kh & 3;
  float ar = 0.0f, ai = 0.0f;
#pragma unroll
  for (int i = 0; i < 8; ++i) {
    size_t zo = ((((size_t)b * 8 + i) * 8 + kd) * 8 + kh) * 4 + kw;
    float xr = ztre[zo], xi = ztim[zo];
    size_t wo = ((((size_t)(i * 8 + o) * 4 + kx) * 4 + ky) * 4 + kw) * 2;  // complex64 interleave
    float wr = w[wo], wi = w[wo + 1];
    ar += xr * wr - xi * wi;
    ai += xr * wi + xi * wr;
  }
  size_t oo = ((((size_t)b * 8 + o) * 8 + kd) * 8 + kh) * 4 + kw;
  ftre[oo] = ar; ftim[oo] = ai;
}

// ---------------------------------------------------------------------------
// Inverse DFT along D: 8 modes -> 64. 131072 threads. Writes into Yd (=Xh buf).
// ---------------------------------------------------------------------------
__global__ __launch_bounds__(256) void k_inv_d(const float* __restrict__ ftre,
                                               const float* __restrict__ ftim,
                                               const float* __restrict__ ifl,
                                               float* __restrict__ ydre,
                                               float* __restrict__ ydim) {
  int t = blockIdx.x * blockDim.x + threadIdx.x;
  int d = t & 63, kw = (t >> 6) & 3, kh = (t >> 8) & 7, bo = t >> 11;
  float ar = 0.0f, ai = 0.0f;
#pragma unroll
  for (int kdi = 0; kdi < 8; ++kdi) {
    float cr = ifl[(kdi * 64 + d) * 2], ci = ifl[(kdi * 64 + d) * 2 + 1];
    size_t fo = (((size_t)bo * 8 + kdi) * 8 + kh) * 4 + kw;
    float xr = ftre[fo], xi = ftim[fo];
    ar += cr * xr - ci * xi;
    ai += cr * xi + ci * xr;
  }
  size_t o = (((size_t)bo * 8 + kh) * 4 + kw) * 64 + d;
  ydre[o] = ar; ydim[o] = ai;
}

// ---------------------------------------------------------------------------
// Inverse DFT along H: 8 modes -> 64. 2^20 threads. Writes into Yh (=Xw buf).
// ---------------------------------------------------------------------------
__global__ __launch_bounds__(256) void k_inv_h(const float* __restrict__ ydre,
                                               const float* __restrict__ ydim,
                                               const float* __restrict__ ifl,
                                               float* __restrict__ yhre,
                                               float* __restrict__ yhim) {
  int t = blockIdx.x * blockDim.x + threadIdx.x;
  int kw = t & 3, h = (t >> 2) & 63, d = (t >> 8) & 63, bo = t >> 14;
  float ar = 0.0f, ai = 0.0f;
#pragma unroll
  for (int khi = 0; khi < 8; ++khi) {
    float cr = ifl[(khi * 64 + h) * 2], ci = ifl[(khi * 64 + h) * 2 + 1];
    size_t yo = (((size_t)bo * 8 + khi) * 4 + kw) * 64 + d;
    float xr = ydre[yo], xi = ydim[yo];
    ar += cr * xr - ci * xi;
    ai += cr * xi + ci * xr;
  }
  size_t o = (((size_t)bo * 64 + d) * 64 + h) * 4 + kw;
  yhre[o] = ar; yhim[o] = ai;
}

// ---------------------------------------------------------------------------
// irfft-W (4 modes -> 64 reals) + pointwise conv + bias + residual + GELU.
// In place on x (each thread reads all i at its point, then writes all o).
// ---------------------------------------------------------------------------
__global__ __launch_bounds__(256) void k_fuse(const float* __restrict__ yhre,
                                              const float* __restrict__ yhim,
                                              const float* __restrict__ iw,
                                              const float* __restrict__ cw,
                                              const float* __restrict__ cb,
                                              const float* xin, float* xout,
                                              int do_gelu) {
  int t = blockIdx.x * blockDim.x + threadIdx.x;
  int w = t & 63;
  int row = t >> 6;                 // b*4096 + d*64 + h
  int b = row >> 12, dh = row & 4095;
  size_t p = (size_t)dh * 64 + w;
  const float* iwt = iw + w * 8;    // [4 k][2]

  float xv[8];
#pragma unroll
  for (int i = 0; i < 8; ++i) xv[i] = xin[(size_t)(b * 8 + i) * 262144 + p];

#pragma unroll
  for (int o = 0; o < 8; ++o) {
    size_t yo = ((size_t)(b * 8 + o) * 4096 + dh) * 4;
    float s = 0.0f;
#pragma unroll
    for (int k = 0; k < 4; ++k)
      s += yhre[yo + k] * iwt[2 * k] + yhim[yo + k] * iwt[2 * k + 1];
    float c = cb[o];
#pragma unroll
    for (int i = 0; i < 8; ++i) c += cw[o * 8 + i] * xv[i];
    float v = s + c;
    if (do_gelu) v = gelu_tanh(v);
    xout[(size_t)(b * 8 + o) * 262144 + p] = v;
  }
}

// ---------------------------------------------------------------------------
// Head: fc1 -> GELU -> fc2, fused per point; plus hard core u*nu, out = hard+0.3*soft
// ---------------------------------------------------------------------------
__global__ __launch_bounds__(256) void k_head(const float* __restrict__ x,
                                              const float* __restrict__ u,
                                              const float* __restrict__ nu,
                                              const float* __restrict__ w1,
                                              const float* __restrict__ b1,
                                              const float* __restrict__ w2,
                                              const float* __restrict__ b2,
                                              float* __restrict__ out) {
  int t = blockIdx.x * blockDim.x + threadIdx.x;  // b*262144 + p
  int p = t & 262143, b = t >> 18;
  float xv[8];
#pragma unroll
  for (int i = 0; i < 8; ++i) xv[i] = x[(size_t)(b * 8 + i) * 262144 + p];
  float hv[32];
#pragma unroll
  for (int j = 0; j < 32; ++j) {
    float a = b1[j];
#pragma unroll
    for (int i = 0; i < 8; ++i) a += w1[j * 8 + i] * xv[i];
    hv[j] = gelu_tanh(a);
  }
  float nuv = nu[0];
#pragma unroll
  for (int oc = 0; oc < 6; ++oc) {
    float s = b2[oc];
#pragma unroll
    for (int j = 0; j < 32; ++j) s += w2[oc * 32 + j] * hv[j];
    float hard = (oc < 3) ? u[(size_t)(b * 3 + oc) * 262144 + p] * nuv : 0.0f;
    out[(size_t)(b * 6 + oc) * 262144 + p] = hard + 0.3f * s;
  }
}

// ---------------------------------------------------------------------------
extern "C" void kernel_launch(void* const* d_in, const int* in_sizes, int n_in,
                              void* d_out, int out_size, void* d_ws, size_t ws_size,
                              hipStream_t stream) {
  float* ws = (float*)d_ws;
  const float* u     = (const float*)d_in[0];
  const float* nu    = (const float*)d_in[1];
  const float* fc0_w = (const float*)d_in[2];
  const float* fc0_b = (const float*)d_in[3];
  const float* spec[16];
  for (int i = 0; i < 16; ++i) spec[i] = (const float*)d_in[4 + i];   // layer-major w1..w4
  const float* convw[4];
  const float* convb[4];
  for (int i = 0; i < 4; ++i) convw[i] = (const float*)d_in[20 + i];
  for (int i = 0; i < 4; ++i) convb[i] = (const float*)d_in[24 + i];
  const float* fc1_w = (const float*)d_in[28];
  const float* fc1_b = (const float*)d_in[29];
  const float* fc2_w = (const float*)d_in[30];
  const float* fc2_b = (const float*)d_in[31];

  float* X    = ws + OFF_X;
  float* XWRE = ws + OFF_XWRE;  // also Yh (re)
  float* XWIM = ws + OFF_XWIM;  // also Yh (im)
  float* XHRE = ws + OFF_XHRE;  // also Yd (re)
  float* XHIM = ws + OFF_XHIM;  // also Yd (im)
  float* ZTRE = ws + OFF_ZTRE;
  float* ZTIM = ws + OFF_ZTIM;
  float* FTRE = ws + OFF_FTRE;
  float* FTIM = ws + OFF_FTIM;

  k_tables<<<14, 256, 0, stream>>>(ws);
  k_fc0<<<8192, 256, 0, stream>>>(u, fc0_w, fc0_b, X);

  for (int L = 0; L < 4; ++L) {
    k_fwd_w<<<4096, 128, 0, stream>>>(X, XWRE, XWIM, ws + OFF_BTAB);
    k_fwd_h<<<512, 256, 0, stream>>>(XWRE, XWIM, ws + OFF_FH, XHRE, XHIM);
    k_fwd_d<<<64, 256, 0, stream>>>(XHRE, XHIM, ws + OFF_FH, ZTRE, ZTIM);
    k_mix<<<64, 256, 0, stream>>>(ZTRE, ZTIM, spec[4 * L + 0], spec[4 * L + 1],
                                  spec[4 * L + 2], spec[4 * L + 3], FTRE, FTIM);
    k_inv_d<<<512, 256, 0, stream>>>(FTRE, FTIM, ws + OFF_IF, XHRE, XHIM);   // Yd = Xh buf
    k_inv_h<<<4096, 256, 0, stream>>>(XHRE, XHIM, ws + OFF_IF, XWRE, XWIM);  // Yh = Xw buf
    k_fuse<<<8192, 256, 0, stream>>>(XWRE, XWIM, ws + OFF_IW, convw[L], convb[L],
                                     X, X, (L < 3) ? 1 : 0);
  }

  k_head<<<8192, 256, 0, stream>>>(X, u, nu, fc1_w, fc1_b, fc2_w, fc2_b, (float*)d_out);
}